// PointCloudExtractor_44564580663678
// MI455X (gfx1250) — compile-verified
//
#include <hip/hip_runtime.h>
#include <hip/hip_bf16.h>

// ---------------- constants ----------------
constexpr int B_   = 8;
constexpr int N_   = 2048;
constexpr int KNB  = 64;      // K neighbors
constexpr int TU_  = 256;
constexpr int U_   = 512;
constexpr int FDIM = 576;     // 3*K*3
constexpr int MR   = B_ * N_; // 16384 rows
constexpr int KC   = 64;      // GEMM K-chunk staged in LDS
constexpr float BNS = 0.9995003746877732f; // 1/sqrt(1+1e-3)

typedef __attribute__((ext_vector_type(2))) float v2f;
typedef __attribute__((ext_vector_type(8))) float v8f;

// ---------------- init ----------------
__global__ void k_init(float* tmax) {
    int i = blockIdx.x * blockDim.x + threadIdx.x;
    if (i < B_ * TU_) tmax[i] = 0.0f;
}

// ---------------- T-Net stage 1: relu(bn(points@tw+tb)) -> max over N ----------------
__global__ __launch_bounds__(256) void k_tnet1(const float* __restrict__ points,
                                               const float* __restrict__ tw,
                                               const float* __restrict__ tb,
                                               const float* __restrict__ tg1,
                                               const float* __restrict__ tbt1,
                                               float* __restrict__ tmax) {
    int u = threadIdx.x;             // 0..255
    int b = blockIdx.y;              // 0..7
    int n0 = blockIdx.x * 64;        // chunk of 64 rows
    float w0 = tw[u], w1 = tw[TU_ + u], w2 = tw[2 * TU_ + u];
    float bb = tb[u];
    float g  = tg1[u] * BNS;
    float bt = tbt1[u];
    float mx = 0.0f;                 // relu >= 0
    const float* P = points + ((size_t)b * N_ + n0) * 3;
    for (int i = 0; i < 64; ++i) {
        float p0 = P[i * 3 + 0], p1 = P[i * 3 + 1], p2 = P[i * 3 + 2];
        float v = fmaf(p0, w0, fmaf(p1, w1, fmaf(p2, w2, bb)));
        v = fmaf(v, g, bt);
        v = fmaxf(v, 0.0f);
        mx = fmaxf(mx, v);
    }
    atomicMax((int*)&tmax[b * TU_ + u], __float_as_int(mx)); // valid: all values >= 0
}

// ---------------- T-Net stage 2: dense->bn->relu, dense -> T[8,3,3] ----------------
__global__ __launch_bounds__(256) void k_tnet2(const float* __restrict__ tmax,
                                               const float* __restrict__ td1w,
                                               const float* __restrict__ td1b,
                                               const float* __restrict__ tg2,
                                               const float* __restrict__ tbt2,
                                               const float* __restrict__ td2w,
                                               const float* __restrict__ td2b,
                                               float* __restrict__ Tmat) {
    __shared__ float h[B_ * TU_];
    int u = threadIdx.x;
    float g  = tg2[u] * BNS;
    float bt = tbt2[u];
    for (int b = 0; b < B_; ++b) {
        float acc = td1b[u];
        const float* x = tmax + b * TU_;
        for (int k = 0; k < TU_; ++k) acc = fmaf(x[k], td1w[k * TU_ + u], acc);
        acc = fmaf(acc, g, bt);
        h[b * TU_ + u] = fmaxf(acc, 0.0f);
    }
    __syncthreads();
    if (u < B_ * 9) {
        int b = u / 9, i = u % 9;
        float acc = td2b[i];
        const float* x = h + b * TU_;
        for (int k = 0; k < TU_; ++k) acc = fmaf(x[k], td2w[k * 9 + i], acc);
        Tmat[b * 9 + i] = acc;
    }
}

// ---------------- transform points: pts = points @ T ----------------
__global__ __launch_bounds__(256) void k_xform(const float* __restrict__ points,
                                               const float* __restrict__ Tmat,
                                               float* __restrict__ pts) {
    int idx = blockIdx.x * blockDim.x + threadIdx.x; // 0..MR-1
    if (idx >= MR) return;
    int b = idx >> 11;
    float p0 = points[idx * 3 + 0], p1 = points[idx * 3 + 1], p2 = points[idx * 3 + 2];
    const float* T = Tmat + b * 9;
    pts[idx * 3 + 0] = fmaf(p0, T[0], fmaf(p1, T[3], p2 * T[6]));
    pts[idx * 3 + 1] = fmaf(p0, T[1], fmaf(p1, T[4], p2 * T[7]));
    pts[idx * 3 + 2] = fmaf(p0, T[2], fmaf(p1, T[5], p2 * T[8]));
}

// ---------------- ball-query top-K + feature build ----------------
// one block per (b,n). Candidates with score>0 almost always < 64, so:
// slots [0..m) = in-ball sorted by noise desc (tie: ascending index, matching
// stable argsort of -score); slots [m..K) = ascending out-of-ball indices.
constexpr int CAP = 256;
__global__ __launch_bounds__(128) void k_ballq(const float* __restrict__ pts,
                                               const float* __restrict__ noise,
                                               float* __restrict__ feats) {
    __shared__ float ptsLoc[N_ * 3];         // 24 KB
    __shared__ float candS[3][CAP];
    __shared__ int   candI[3][CAP];
    __shared__ int   cnt[3];
    __shared__ int   slot[3][KNB];

    const float rad2[3] = {0.1f * 0.1f, 0.3f * 0.3f, 0.7f * 0.7f};
    int tid = threadIdx.x;
    int b = blockIdx.x >> 11;
    int n = blockIdx.x & (N_ - 1);

    for (int i = tid; i < N_ * 3; i += blockDim.x)
        ptsLoc[i] = pts[(size_t)b * N_ * 3 + i];
    if (tid < 3) cnt[tid] = 0;
    __syncthreads();

    float qx = ptsLoc[n * 3 + 0], qy = ptsLoc[n * 3 + 1], qz = ptsLoc[n * 3 + 2];
    const float* nrow = noise + ((size_t)b * N_ + n) * N_;

    for (int j = tid; j < N_; j += blockDim.x) {
        float dx = qx - ptsLoc[j * 3 + 0];
        float dy = qy - ptsLoc[j * 3 + 1];
        float dz = qz - ptsLoc[j * 3 + 2];
        float d2 = fmaf(dx, dx, fmaf(dy, dy, dz * dz));
        if (d2 <= rad2[2]) {                     // inside largest ball: touch noise once
            float sc = nrow[j];
            #pragma unroll
            for (int r = 0; r < 3; ++r) {
                if (d2 <= rad2[r]) {
                    int pos = atomicAdd(&cnt[r], 1);
                    if (pos < CAP) { candS[r][pos] = sc; candI[r][pos] = j; }
                }
            }
        }
    }
    __syncthreads();

    // rank-sort in-ball candidates (descending score, stable by index)
    for (int r = 0; r < 3; ++r) {
        int m = cnt[r]; if (m > CAP) m = CAP;
        for (int i = tid; i < m; i += blockDim.x) {
            float si = candS[r][i]; int ii = candI[r][i];
            int rank = 0;
            for (int q = 0; q < m; ++q) {
                float sq = candS[r][q];
                rank += (sq > si) || (sq == si && candI[r][q] < ii);
            }
            if (rank < KNB) slot[r][rank] = ii;
        }
    }
    // pad with ascending out-of-ball indices (zero-score entries, stable order)
    if (tid < 3) {
        int r = tid;
        int m = cnt[r]; if (m > CAP) m = CAP;
        int c = m;
        for (int j = 0; c < KNB && j < N_; ++j) {
            float dx = qx - ptsLoc[j * 3 + 0];
            float dy = qy - ptsLoc[j * 3 + 1];
            float dz = qz - ptsLoc[j * 3 + 2];
            float d2 = fmaf(dx, dx, fmaf(dy, dy, dz * dz));
            if (d2 > rad2[r]) slot[r][c++] = j;
        }
    }
    __syncthreads();

    float* out = feats + (size_t)blockIdx.x * FDIM;
    for (int t = tid; t < FDIM; t += blockDim.x) {
        int k = t / 9, rr = (t % 9) / 3, c = t % 3;
        out[t] = ptsLoc[slot[rr][k] * 3 + c];
    }
}

// ---------------- WMMA fp32 GEMM + fused epilogue, LDS-staged ----------------
// Block: 256 threads = 8 waves = 4 m-subtiles x 2 n-subtiles -> 64x32 output tile.
// K staged in LDS in chunks of KC=64; B stored transposed so each fragment is one
// aligned ds_load_b64. All K dims (576/512/256) are multiples of KC.
// mode 0: Out = relu((X@W + bias) * gamma*BNS + beta)
// mode 1: Out = resid + X@W + bias
__global__ __launch_bounds__(256) void k_gemm_wmma(const float* __restrict__ X,
                                                   const float* __restrict__ W,
                                                   const float* __restrict__ bias,
                                                   const float* __restrict__ gamma,
                                                   const float* __restrict__ beta,
                                                   const float* __restrict__ resid,
                                                   float* __restrict__ Out,
                                                   int Kdim, int Ncols, int mode) {
    __shared__ float lds_a[64][KC];   // 16 KB : rows x k
    __shared__ float lds_b[32][KC];   // 8 KB  : cols x k (transposed)

    const int tid  = threadIdx.x;
    const int wave = tid >> 5;
    const int lane = tid & 31;
    const int wm   = wave >> 1;       // 0..3
    const int wn   = wave & 1;        // 0..1
    const int ntiles = (Ncols + 15) >> 4;
    const int tn   = blockIdx.y * 2 + wn;           // 16-col subtile index
    const bool waveActive = (tn < ntiles);          // wave-uniform

    const int m0 = blockIdx.x * 64;                 // block row base
    const int fr = lane & 15;                       // fragment row/col within 16
    const int khalf = (lane >> 4) << 1;             // K sub-pair 0 or 2
    const bool vec4 = (Ncols & 3) == 0;

    v8f acc = {};
    for (int kb0 = 0; kb0 < Kdim; kb0 += KC) {
        __syncthreads();
        // ---- stage A: 64 rows x KC, coalesced float4 ----
        #pragma unroll
        for (int i = 0; i < 4; ++i) {
            int idx = tid + i * 256;                // 0..1023 float4 slots
            int r   = idx >> 4;                     // KC/4 = 16 float4 per row
            int k4  = (idx & 15) << 2;
            float4 v = *(const float4*)&X[(size_t)(m0 + r) * Kdim + kb0 + k4];
            lds_a[r][k4 + 0] = v.x; lds_a[r][k4 + 1] = v.y;
            lds_a[r][k4 + 2] = v.z; lds_a[r][k4 + 3] = v.w;
        }
        // ---- stage B: KC rows x 32 cols, transposed into [col][k] ----
        #pragma unroll
        for (int i = 0; i < 2; ++i) {
            int idx = tid + i * 256;                // 0..511 float4 slots
            int kr  = idx >> 3;                     // 8 float4 per 32-col row
            int c4  = (idx & 7) << 2;
            int c0  = blockIdx.y * 32 + c4;
            const float* wrow = W + (size_t)(kb0 + kr) * Ncols;
            if (vec4 && c0 + 3 < Ncols) {
                float4 v = *(const float4*)&wrow[c0];
                lds_b[c4 + 0][kr] = v.x; lds_b[c4 + 1][kr] = v.y;
                lds_b[c4 + 2][kr] = v.z; lds_b[c4 + 3][kr] = v.w;
            } else {
                #pragma unroll
                for (int j = 0; j < 4; ++j) {
                    int cc = c0 + j; if (cc > Ncols - 1) cc = Ncols - 1; // clamp: garbage ok
                    lds_b[c4 + j][kr] = wrow[cc];
                }
            }
        }
        __syncthreads();
        // ---- 16 WMMAs per chunk, all operands from LDS ----
        if (waveActive) {
            const float* arow = &lds_a[wm * 16 + fr][khalf];
            const float* brow = &lds_b[wn * 16 + fr][khalf];
            #pragma unroll
            for (int kk = 0; kk < KC; kk += 4) {
                v2f a = *(const v2f*)(arow + kk);
                v2f bf = *(const v2f*)(brow + kk);
                acc = __builtin_amdgcn_wmma_f32_16x16x4_f32(
                    false, a, false, bf, (short)0, acc, false, false);
            }
        }
    }

    int colD = tn * 16 + fr;
    if (!waveActive || colD >= Ncols) return;      // after all barriers/WMMA
    float bi = bias[colD];
    float g = 0.0f, be = 0.0f;
    if (mode == 0) { g = gamma[colD] * BNS; be = beta[colD]; }
    int rbase = m0 + wm * 16 + ((lane >> 4) << 3); // C/D: VGPR q -> row q (+8 upper half)
    #pragma unroll
    for (int q = 0; q < 8; ++q) {
        int r = rbase + q;
        float v = acc[q] + bi;
        if (mode == 0) {
            v = fmaxf(fmaf(v, g, be), 0.0f);
        } else {
            v += resid[(size_t)r * Ncols + colD];
        }
        Out[(size_t)r * Ncols + colD] = v;
    }
}

// ---------------- column-wise max over N ----------------
__global__ __launch_bounds__(256) void k_colmax(const float* __restrict__ f,
                                                float* __restrict__ out) {
    __shared__ float red[256];
    int u = blockIdx.x;   // 0..169
    int b = blockIdx.y;   // 0..7
    float mx = 0.0f;      // relu >= 0
    for (int n = threadIdx.x; n < N_; n += 256)
        mx = fmaxf(mx, f[((size_t)b * N_ + n) * 170 + u]);
    red[threadIdx.x] = mx;
    __syncthreads();
    for (int s = 128; s > 0; s >>= 1) {
        if (threadIdx.x < s) red[threadIdx.x] = fmaxf(red[threadIdx.x], red[threadIdx.x + s]);
        __syncthreads();
    }
    if (threadIdx.x == 0) out[b * 170 + u] = red[0];
}

// ---------------- launch ----------------
extern "C" void kernel_launch(void* const* d_in, const int* in_sizes, int n_in,
                              void* d_out, int out_size, void* d_ws, size_t ws_size,
                              hipStream_t stream) {
    const float* points = (const float*)d_in[0];
    const float* noise  = (const float*)d_in[1];
    const float* tw     = (const float*)d_in[2];
    const float* tb     = (const float*)d_in[3];
    const float* tg1    = (const float*)d_in[4];
    const float* tbt1   = (const float*)d_in[5];
    const float* td1w   = (const float*)d_in[6];
    const float* td1b   = (const float*)d_in[7];
    const float* tg2    = (const float*)d_in[8];
    const float* tbt2   = (const float*)d_in[9];
    const float* td2w   = (const float*)d_in[10];
    const float* td2b   = (const float*)d_in[11];
    const float* c1w    = (const float*)d_in[12];
    const float* c1b    = (const float*)d_in[13];
    const float* g1     = (const float*)d_in[14];
    const float* be1    = (const float*)d_in[15];
    const float* rw     = (const float*)d_in[16];
    const float* rb     = (const float*)d_in[17];
    const float* b1w = (const float*)d_in[18], *b1b = (const float*)d_in[19];
    const float* b1g = (const float*)d_in[20], *b1be = (const float*)d_in[21];
    const float* b2w = (const float*)d_in[22], *b2b = (const float*)d_in[23];
    const float* b2g = (const float*)d_in[24], *b2be = (const float*)d_in[25];
    const float* b3w = (const float*)d_in[26], *b3b = (const float*)d_in[27];
    const float* b3g = (const float*)d_in[28], *b3be = (const float*)d_in[29];
    float* out = (float*)d_out;

    float* ws    = (float*)d_ws;
    float* tmax  = ws;                          // 2048
    float* Tmat  = tmax + B_ * TU_;             // 72 (pad 80)
    float* pts   = Tmat + 80;                   // 49152
    float* feats = pts + (size_t)MR * 3;        // MR*576 (also reused as ping buffer)
    float* buf   = feats + (size_t)MR * FDIM;   // MR*512

    k_init<<<(B_ * TU_ + 255) / 256, 256, 0, stream>>>(tmax);
    k_tnet1<<<dim3(N_ / 64, B_), 256, 0, stream>>>(points, tw, tb, tg1, tbt1, tmax);
    k_tnet2<<<1, 256, 0, stream>>>(tmax, td1w, td1b, tg2, tbt2, td2w, td2b, Tmat);
    k_xform<<<MR / 256, 256, 0, stream>>>(points, Tmat, pts);
    k_ballq<<<MR, 128, 0, stream>>>(pts, noise, feats);

    // L1: feats[M,576] @ c1w[576,512] -> bn/relu -> buf
    k_gemm_wmma<<<dim3(MR / 64, 16), 256, 0, stream>>>(
        feats, c1w, c1b, g1, be1, nullptr, buf, FDIM, U_, 0);
    // L2: f1 = f0 + (f0 @ rw + rb) -> feats(region, stride 512)
    k_gemm_wmma<<<dim3(MR / 64, 16), 256, 0, stream>>>(
        buf, rw, rb, nullptr, nullptr, buf, feats, U_, U_, 1);
    // L3: f1 @ b1w[512,512] -> bn/relu -> buf
    k_gemm_wmma<<<dim3(MR / 64, 16), 256, 0, stream>>>(
        feats, b1w, b1b, b1g, b1be, nullptr, buf, U_, U_, 0);
    // L4: f2 @ b2w[512,256] -> bn/relu -> feats(region, stride 256)
    k_gemm_wmma<<<dim3(MR / 64, 8), 256, 0, stream>>>(
        buf, b2w, b2b, b2g, b2be, nullptr, feats, U_, 256, 0);
    // L5: f3 @ b3w[256,170] -> bn/relu -> buf(stride 170); 11 n-subtiles -> grid.y=6
    k_gemm_wmma<<<dim3(MR / 64, 6), 256, 0, stream>>>(
        feats, b3w, b3b, b3g, b3be, nullptr, buf, 256, 170, 0);

    k_colmax<<<dim3(170, B_), 256, 0, stream>>>(buf, out);
}